// Quantizer_61804579389567
// MI455X (gfx1250) — compile-verified
//
#include <hip/hip_runtime.h>
#include <stdint.h>

// Quantize + bit-error-injection + dequantize for MI455X (gfx1250).
// Pass 1: global absmax reduction (wave32 shfl + LDS + global_atomic_max_u32).
// Pass 2: streaming elementwise kernel; noise_u streamed via async
//         global->LDS DMA (ASYNCcnt) with double buffering, NT hints so the
//         256MB noise stream does not evict x (64MB, L2-resident from pass 1).
// Roofline: ~384MB effective HBM traffic @ 23.3 TB/s -> ~16.5us; VALU is slack.

typedef __attribute__((ext_vector_type(4))) float v4f;
typedef __attribute__((ext_vector_type(4))) int   v4i;

#define TILE 1024  // elements per block-tile: 256 threads * 4 elems

#if defined(__gfx1250__) &&                                                  \
    __has_builtin(__builtin_amdgcn_global_load_async_to_lds_b128) &&         \
    __has_builtin(__builtin_amdgcn_s_wait_asynccnt)
#define USE_ASYNC 1
// Builtin prototype (from hipcc diagnostic): (v4i as1*, v4i as3*, Iint, Iint)
typedef __attribute__((address_space(1))) v4i* gbl_v4i_ptr;
typedef __attribute__((address_space(3))) v4i* lds_v4i_ptr;
#else
#define USE_ASYNC 0
#endif

__global__ void __launch_bounds__(1) ws_init_kernel(unsigned* ws) {
  ws[0] = 0u;  // absmax accumulator (uint bit-pattern of non-negative float)
}

__global__ void __launch_bounds__(256)
absmax_kernel(const float* __restrict__ x, unsigned* __restrict__ ws, long long n) {
  const long long n4  = n >> 2;
  const long long gid = (long long)blockIdx.x * blockDim.x + threadIdx.x;
  const long long gs  = (long long)gridDim.x * blockDim.x;
  const v4f* __restrict__ xv = (const v4f*)x;

  float m = 0.0f;
  for (long long i = gid; i < n4; i += gs) {
    v4f v = xv[i];
    m = fmaxf(fmaxf(fabsf(v.x), fabsf(v.y)),
              fmaxf(fmaxf(fabsf(v.z), fabsf(v.w)), m));
  }
  if (gid == 0) {  // scalar tail (n % 4)
    for (long long i = n4 << 2; i < n; ++i) m = fmaxf(m, fabsf(x[i]));
  }

  // wave32 butterfly reduction
  #pragma unroll
  for (int off = 16; off > 0; off >>= 1)
    m = fmaxf(m, __shfl_xor(m, off, 32));

  __shared__ float smax[8];  // 256 threads = 8 waves (wave32)
  const int lane = threadIdx.x & 31;
  const int wave = threadIdx.x >> 5;
  if (lane == 0) smax[wave] = m;
  __syncthreads();
  if (threadIdx.x == 0) {
    float t = smax[0];
    #pragma unroll
    for (int i = 1; i < 8; ++i) t = fmaxf(t, smax[i]);
    // abs floats compare identically as uints -> global_atomic_max_u32
    atomicMax(ws, __float_as_uint(t));
  }
}

__device__ __forceinline__ float qnoise_elem(float xv, v4f nv, float scale) {
  float xs = xv / scale;                       // IEEE div, matches x / scale
  xs = fminf(fmaxf(xs, -8.0f), 7.0f);          // clip(x_s, Qn, Qp)
  const int qi = (int)rintf(xs);               // round-half-even == jnp.round
  const unsigned u = ((unsigned)qi) & 0xFu;    // signed -> 4-bit two's compl.
  const unsigned eps = (nv.x < 0.01f ? 1u : 0u) | (nv.y < 0.01f ? 2u : 0u) |
                       (nv.z < 0.01f ? 4u : 0u) | (nv.w < 0.01f ? 8u : 0u);
  const unsigned up = u ^ eps;
  const int p = (int)(up & 7u) - (int)(up & 8u);  // bin2int sign extend
  return (float)p * scale;
}

__global__ void __launch_bounds__(256)
qnoise_kernel(const float* __restrict__ x, const float* __restrict__ noise,
              float* __restrict__ out, const unsigned* __restrict__ ws,
              long long n) {
  const float scale = __uint_as_float(ws[0]) / 7.0f;
  const v4f* __restrict__ nu = (const v4f*)noise;
  const int tid = threadIdx.x;
  const long long nFull = n & ~(long long)(TILE - 1);
  const long long tileStride = (long long)gridDim.x * TILE;

#if USE_ASYNC
  __shared__ v4f nbuf[2][TILE];  // 32 KB double buffer for the noise stream
  const long long t0 = (long long)blockIdx.x * TILE;
  int buf = 0;
  if (t0 < nFull) {  // prologue: DMA tile 0 into buffer 0
    const v4f* g = nu + t0 + 4 * tid;
    v4f* l = &nbuf[0][4 * tid];
    #pragma unroll
    for (int k = 0; k < 4; ++k)
      __builtin_amdgcn_global_load_async_to_lds_b128(
          (gbl_v4i_ptr)(g + k), (lds_v4i_ptr)(l + k), 0, 0);
  }
  for (long long t = t0; t < nFull; t += tileStride) {
    const long long nxt = t + tileStride;
    if (nxt < nFull) {  // DMA next tile while current computes
      const v4f* g = nu + nxt + 4 * tid;
      v4f* l = &nbuf[buf ^ 1][4 * tid];
      #pragma unroll
      for (int k = 0; k < 4; ++k)
        __builtin_amdgcn_global_load_async_to_lds_b128(
            (gbl_v4i_ptr)(g + k), (lds_v4i_ptr)(l + k), 0, 0);
      __builtin_amdgcn_s_wait_asynccnt(4);  // in-order: current tile landed
    } else {
      __builtin_amdgcn_s_wait_asynccnt(0);
    }
    const long long e = t + 4 * (long long)tid;
    const v4f xv = *(const v4f*)(x + e);  // RT hint: x stays L2-resident
    v4f o;
    o.x = qnoise_elem(xv.x, nbuf[buf][4 * tid + 0], scale);
    o.y = qnoise_elem(xv.y, nbuf[buf][4 * tid + 1], scale);
    o.z = qnoise_elem(xv.z, nbuf[buf][4 * tid + 2], scale);
    o.w = qnoise_elem(xv.w, nbuf[buf][4 * tid + 3], scale);
    __builtin_nontemporal_store(o, (v4f*)(out + e));  // streamed output
    buf ^= 1;
  }
#else
  for (long long t = (long long)blockIdx.x * TILE; t < nFull; t += tileStride) {
    const long long e = t + 4 * (long long)tid;
    const v4f xv = *(const v4f*)(x + e);
    const v4f n0 = __builtin_nontemporal_load(nu + e + 0);
    const v4f n1 = __builtin_nontemporal_load(nu + e + 1);
    const v4f n2 = __builtin_nontemporal_load(nu + e + 2);
    const v4f n3 = __builtin_nontemporal_load(nu + e + 3);
    v4f o;
    o.x = qnoise_elem(xv.x, n0, scale);
    o.y = qnoise_elem(xv.y, n1, scale);
    o.z = qnoise_elem(xv.z, n2, scale);
    o.w = qnoise_elem(xv.w, n3, scale);
    __builtin_nontemporal_store(o, (v4f*)(out + e));
  }
#endif

  // scalar tail for n not a multiple of TILE (not hit for 4096x4096)
  const long long gid = (long long)blockIdx.x * blockDim.x + tid;
  const long long gs  = (long long)gridDim.x * blockDim.x;
  for (long long i = nFull + gid; i < n; i += gs)
    out[i] = qnoise_elem(x[i], nu[i], scale);
}

extern "C" void kernel_launch(void* const* d_in, const int* in_sizes, int n_in,
                              void* d_out, int out_size, void* d_ws, size_t ws_size,
                              hipStream_t stream) {
  const float* x     = (const float*)d_in[0];
  const float* noise = (const float*)d_in[1];
  float* out         = (float*)d_out;
  unsigned* ws       = (unsigned*)d_ws;
  const long long n  = (long long)in_sizes[0];  // 4096*4096

  ws_init_kernel<<<1, 1, 0, stream>>>(ws);
  absmax_kernel<<<2048, 256, 0, stream>>>(x, ws, n);
  qnoise_kernel<<<2048, 256, 0, stream>>>(x, noise, out, ws, n);
}